// MemNet_66151086293668
// MI455X (gfx1250) — compile-verified
//
#include <hip/hip_runtime.h>
#include <hip/hip_bf16.h>

// ---------------------------------------------------------------------------
// MemNet memory-attention for MI455X (gfx1250), wave32 + V_WMMA_F32_16X16X4_F32
//
// Math notes (why some inputs are ignored):
//  * mask m(b,h,w) in {1, 1e-7} multiplies feat BEFORE channel L2-normalize,
//    and normalize(m*x)=normalize(x) for m>0  => target_mask / patch_size are
//    no-ops on the output; qf is shared by the real & fake passes.
//  * colmax[m] = max_n softmax_col(score) = 1/colsum[m]  =>
//    w[n] = exp(score[n,g] - colmax_pre[g]), g = argmax_m score[n,m].
//
// GEMM inner loops are 2-deep register double-buffered so v_wmma consumes
// fragments loaded two k-steps earlier (waits tolerate ~2 iterations of
// memory latency instead of serializing load->wmma).
// ---------------------------------------------------------------------------

typedef float v2f __attribute__((ext_vector_type(2)));
typedef float v8f __attribute__((ext_vector_type(8)));

static constexpr int Bc  = 8;
static constexpr int Cc  = 512;
static constexpr int Hh  = 64;
static constexpr int Ww  = 64;
static constexpr int HW  = Hh * Ww;          // 4096
static constexpr int Nn  = Bc * HW;          // 32768 queries
static constexpr int Mm  = 1024;             // memory slots
static constexpr int OUTC = 4 * Cc;          // 2048 output channels
static constexpr int NCHUNK = 32;            // column-max partial chunks

#define WMMA_F32(A, B, C) \
  __builtin_amdgcn_wmma_f32_16x16x4_f32(false, (A), false, (B), (short)0, (C), false, false)

// ---------------------------------------------------------------------------
// qf = l2norm(feat over channels), layout [N, C]; also write qf into output
// channel blocks [0,512) (real half) and [1024,1536) (fake half).
// ---------------------------------------------------------------------------
__global__ void __launch_bounds__(128)
k_qf_norm(const float* __restrict__ feat, float* __restrict__ qf,
          float* __restrict__ out) {
  const int n  = blockIdx.x;
  const int b  = n >> 12;           // / HW
  const int hw = n & (HW - 1);
  const int t  = threadIdx.x;
  const float* src = feat + (size_t)b * Cc * HW + hw;   // channel stride = HW

  float v[4];
  float ss = 0.f;
#pragma unroll
  for (int i = 0; i < 4; ++i) {
    const int c = t + i * 128;
    v[i] = src[(size_t)c * HW];
    ss += v[i] * v[i];
  }
  __shared__ float red[128];
  red[t] = ss;
  __syncthreads();
  for (int s = 64; s > 0; s >>= 1) {
    if (t < s) red[t] += red[t + s];
    __syncthreads();
  }
  const float inv = 1.0f / fmaxf(sqrtf(red[0]), 1e-12f);

  float* outb = out + (size_t)b * OUTC * HW + hw;
#pragma unroll
  for (int i = 0; i < 4; ++i) {
    const int c = t + i * 128;
    const float q = v[i] * inv;
    qf[(size_t)n * Cc + c] = q;
    outb[(size_t)c * HW]          = q;   // channels [0,512): real-pass qf
    outb[(size_t)(c + 1024) * HW] = q;   // channels [1024,1536): fake-pass qf
  }
}

__global__ void k_zero(float* __restrict__ p, int n) {
  const int i = blockIdx.x * 256 + threadIdx.x;
  if (i < n) p[i] = 0.f;
}

// ---------------------------------------------------------------------------
// Cout[N,Mc] = A[N,K] @ B[Mc,K]^T with V_WMMA_F32_16X16X4_F32.
// Wave tile: 16 (n) x 64 (m) = 4 accumulators sharing one A fragment.
// Block = 128 threads (4 waves) -> 16 x 256 block tile.
// ---------------------------------------------------------------------------
struct FragNT { v2f a, b0, b1, b2, b3; };

__device__ __forceinline__ FragNT ld_nt(const float* ap, const float* b0,
                                        const float* b1, const float* b2,
                                        const float* b3, int k) {
  FragNT f;
  f.a  = *(const v2f*)(ap + k);
  f.b0 = *(const v2f*)(b0 + k);
  f.b1 = *(const v2f*)(b1 + k);
  f.b2 = *(const v2f*)(b2 + k);
  f.b3 = *(const v2f*)(b3 + k);
  return f;
}

__global__ void __launch_bounds__(128)
k_gemm_ab_t(const float* __restrict__ A, const float* __restrict__ B,
            float* __restrict__ Cout, int K, int Mc) {
  const int lane = threadIdx.x & 31;
  const int wv   = threadIdx.x >> 5;
  const int tn   = blockIdx.x << 4;
  const int tm   = (blockIdx.y << 8) + (wv << 6);
  const int lr   = lane & 15;            // row within fragment
  const int lk   = (lane >> 4) << 1;     // K pair select (ISA 16x4 A layout)

  const float* ap = A + (size_t)(tn + lr) * K + lk;
  const float* b0 = B + (size_t)(tm + lr) * K + lk;
  const float* b1 = b0 + (size_t)16 * K;
  const float* b2 = b0 + (size_t)32 * K;
  const float* b3 = b0 + (size_t)48 * K;

  v8f acc0 = {}, acc1 = {}, acc2 = {}, acc3 = {};

  // 2-deep software pipeline: WMMA consumes fragments loaded 2 k-steps ago.
  FragNT cur = ld_nt(ap, b0, b1, b2, b3, 0);
  FragNT nxt = ld_nt(ap, b0, b1, b2, b3, 4);
#pragma unroll 4
  for (int k = 0; k + 8 < K; k += 4) {
    __builtin_prefetch(ap + k + 64, 0, 3);   // global_prefetch_b8 (near)
    __builtin_prefetch(b0 + k + 64, 0, 3);
    FragNT fut = ld_nt(ap, b0, b1, b2, b3, k + 8);
    acc0 = WMMA_F32(cur.a, cur.b0, acc0);
    acc1 = WMMA_F32(cur.a, cur.b1, acc1);
    acc2 = WMMA_F32(cur.a, cur.b2, acc2);
    acc3 = WMMA_F32(cur.a, cur.b3, acc3);
    cur = nxt;
    nxt = fut;
  }
  acc0 = WMMA_F32(cur.a, cur.b0, acc0);
  acc1 = WMMA_F32(cur.a, cur.b1, acc1);
  acc2 = WMMA_F32(cur.a, cur.b2, acc2);
  acc3 = WMMA_F32(cur.a, cur.b3, acc3);
  acc0 = WMMA_F32(nxt.a, nxt.b0, acc0);
  acc1 = WMMA_F32(nxt.a, nxt.b1, acc1);
  acc2 = WMMA_F32(nxt.a, nxt.b2, acc2);
  acc3 = WMMA_F32(nxt.a, nxt.b3, acc3);

  const int rv = (lane >> 4) << 3;       // C/D: lanes>=16 hold M=v+8
#pragma unroll
  for (int v = 0; v < 8; ++v) {
    const size_t row = (size_t)(tn + rv + v);
    float* crow = Cout + row * Mc + tm + lr;
    crow[0]  = acc0[v];
    crow[16] = acc1[v];
    crow[32] = acc2[v];
    crow[48] = acc3[v];
  }
}

// ---------------------------------------------------------------------------
// Column max of score [N,M] -> colmax_pre[M] (two stages, coalesced).
// ---------------------------------------------------------------------------
__global__ void __launch_bounds__(256)
k_colmax_part(const float* __restrict__ S, float* __restrict__ pmax, int rowsPer) {
  const int m  = blockIdx.x * 256 + threadIdx.x;
  const int r0 = blockIdx.y * rowsPer;
  float mx = -3.4e38f;
  for (int r = 0; r < rowsPer; ++r)
    mx = fmaxf(mx, S[(size_t)(r0 + r) * Mm + m]);
  pmax[(size_t)blockIdx.y * Mm + m] = mx;
}

__global__ void __launch_bounds__(256)
k_colmax_fin(const float* __restrict__ pmax, float* __restrict__ colmax) {
  const int m = blockIdx.x * 256 + threadIdx.x;
  float mx = -3.4e38f;
#pragma unroll
  for (int c = 0; c < NCHUNK; ++c)
    mx = fmaxf(mx, pmax[(size_t)c * Mm + m]);
  colmax[m] = mx;
}

// ---------------------------------------------------------------------------
// Per row n: g = argmax_m score, w = exp(score[n,g]-colmax[g]);
// q_update[g,:] += w * qf[n,:]  (float atomics -> global_atomic_add_f32).
// ---------------------------------------------------------------------------
__global__ void __launch_bounds__(256)
k_rowpick_update(const float* __restrict__ S, const float* __restrict__ colmax,
                 const float* __restrict__ qf, float* __restrict__ qup) {
  const int n = blockIdx.x;
  const int t = threadIdx.x;
  const float* row = S + (size_t)n * Mm;

  float bv = -3.4e38f; int bi = 0;
#pragma unroll
  for (int i = 0; i < 4; ++i) {
    const int m = t + i * 256;
    const float v = row[m];
    if (v > bv) { bv = v; bi = m; }
  }
  __shared__ float sv[256];
  __shared__ int   si[256];
  sv[t] = bv; si[t] = bi;
  __syncthreads();
  for (int s = 128; s > 0; s >>= 1) {
    if (t < s && sv[t + s] > sv[t]) { sv[t] = sv[t + s]; si[t] = si[t + s]; }
    __syncthreads();
  }
  __shared__ float wsh;
  __shared__ int   gsh;
  if (t == 0) {
    const int g = si[0];
    gsh = g;
    wsh = __expf(sv[0] - colmax[g]);
  }
  __syncthreads();
  const int g = gsh;
  const float w = wsh;
  const float* qrow = qf + (size_t)n * Cc;
  float* urow = qup + (size_t)g * Cc;
  atomicAdd(&urow[t],       w * qrow[t]);
  atomicAdd(&urow[t + 256], w * qrow[t + 256]);
}

// mem_upd[m,:] = l2norm(q_update[m,:] + mem[m,:])
__global__ void __launch_bounds__(256)
k_memupd_norm(const float* __restrict__ qup, const float* __restrict__ mem,
              float* __restrict__ memupd) {
  const int m = blockIdx.x;
  const int t = threadIdx.x;
  const size_t base = (size_t)m * Cc;
  const float a0 = qup[base + t]       + mem[base + t];
  const float a1 = qup[base + t + 256] + mem[base + t + 256];
  __shared__ float red[256];
  red[t] = a0 * a0 + a1 * a1;
  __syncthreads();
  for (int s = 128; s > 0; s >>= 1) {
    if (t < s) red[t] += red[t + s];
    __syncthreads();
  }
  const float inv = 1.0f / fmaxf(sqrtf(red[0]), 1e-12f);
  memupd[base + t]       = a0 * inv;
  memupd[base + t + 256] = a1 * inv;
}

// Per row of score2: rowmax and 1/sum(exp(x-rowmax)).
__global__ void __launch_bounds__(256)
k_rowstats(const float* __restrict__ S, float* __restrict__ rmax,
           float* __restrict__ rinv) {
  const int n = blockIdx.x;
  const int t = threadIdx.x;
  const float* row = S + (size_t)n * Mm;
  float v[4];
  float mx = -3.4e38f;
#pragma unroll
  for (int i = 0; i < 4; ++i) {
    v[i] = row[t + i * 256];
    mx = fmaxf(mx, v[i]);
  }
  __shared__ float red[256];
  red[t] = mx;
  __syncthreads();
  for (int s = 128; s > 0; s >>= 1) {
    if (t < s) red[t] = fmaxf(red[t], red[t + s]);
    __syncthreads();
  }
  const float rowMax = red[0];
  __syncthreads();                       // before overwriting red[]
  float sum = 0.f;
#pragma unroll
  for (int i = 0; i < 4; ++i) sum += __expf(v[i] - rowMax);
  red[t] = sum;
  __syncthreads();
  for (int s = 128; s > 0; s >>= 1) {
    if (t < s) red[t] += red[t + s];
    __syncthreads();
  }
  if (t == 0) {
    rmax[n] = rowMax;
    rinv[n] = 1.0f / red[0];
  }
}

// ---------------------------------------------------------------------------
// out[:, chanBase..chanBase+512) = softmax_row(score2) @ mem_upd, fused:
// A = exp(score2 - rowmax) on the fly, row scale 1/rowsum applied at store.
// Output written transposed straight into [B, 2048, H, W].
// ---------------------------------------------------------------------------
struct FragNN { v2f a; float b0x, b0y, b1x, b1y, b2x, b2y, b3x, b3y; };

__device__ __forceinline__ FragNN ld_nn(const float* ap, const float* bp, int k) {
  FragNN f;
  f.a = *(const v2f*)(ap + k);
  const float* bk = bp + (size_t)k * Cc;
  f.b0x = bk[0];  f.b0y = bk[Cc];
  f.b1x = bk[16]; f.b1y = bk[Cc + 16];
  f.b2x = bk[32]; f.b2y = bk[Cc + 32];
  f.b3x = bk[48]; f.b3y = bk[Cc + 48];
  return f;
}

__global__ void __launch_bounds__(128)
k_softmax_gemm_out(const float* __restrict__ S, const float* __restrict__ Bm,
                   const float* __restrict__ rmax, const float* __restrict__ rinv,
                   float* __restrict__ out, int chanBase) {
  const int lane = threadIdx.x & 31;
  const int wv   = threadIdx.x >> 5;
  const int tn   = blockIdx.x << 4;
  const int tc   = (blockIdx.y << 8) + (wv << 6);
  const int lr   = lane & 15;
  const int lk   = (lane >> 4) << 1;

  const int arow = tn + lr;
  const float rm = rmax[arow];
  const float* ap = S + (size_t)arow * Mm + lk;
  const float* bp = Bm + (size_t)lk * Cc + tc + lr;

  v8f acc0 = {}, acc1 = {}, acc2 = {}, acc3 = {};

  FragNN cur = ld_nn(ap, bp, 0);
  FragNN nxt = ld_nn(ap, bp, 4);
#pragma unroll 2
  for (int k = 0; k + 8 < Mm; k += 4) {
    FragNN fut = ld_nn(ap, bp, k + 8);
    v2f af;  af.x = __expf(cur.a.x - rm);  af.y = __expf(cur.a.y - rm);
    v2f bf0; bf0.x = cur.b0x; bf0.y = cur.b0y;
    v2f bf1; bf1.x = cur.b1x; bf1.y = cur.b1y;
    v2f bf2; bf2.x = cur.b2x; bf2.y = cur.b2y;
    v2f bf3; bf3.x = cur.b3x; bf3.y = cur.b3y;
    acc0 = WMMA_F32(af, bf0, acc0);
    acc1 = WMMA_F32(af, bf1, acc1);
    acc2 = WMMA_F32(af, bf2, acc2);
    acc3 = WMMA_F32(af, bf3, acc3);
    cur = nxt;
    nxt = fut;
  }
#pragma unroll
  for (int e = 0; e < 2; ++e) {
    const FragNN& f = e ? nxt : cur;
    v2f af;  af.x = __expf(f.a.x - rm);  af.y = __expf(f.a.y - rm);
    v2f bf0; bf0.x = f.b0x; bf0.y = f.b0y;
    v2f bf1; bf1.x = f.b1x; bf1.y = f.b1y;
    v2f bf2; bf2.x = f.b2x; bf2.y = f.b2y;
    v2f bf3; bf3.x = f.b3x; bf3.y = f.b3y;
    acc0 = WMMA_F32(af, bf0, acc0);
    acc1 = WMMA_F32(af, bf1, acc1);
    acc2 = WMMA_F32(af, bf2, acc2);
    acc3 = WMMA_F32(af, bf3, acc3);
  }

  const int rv = (lane >> 4) << 3;
  const int c0 = tc + lr;
#pragma unroll
  for (int v = 0; v < 8; ++v) {
    const int n  = tn + rv + v;
    const float sc = rinv[n];
    const int b  = n >> 12;
    const int hw = n & (HW - 1);
    float* ob = out + (size_t)b * OUTC * HW + (size_t)chanBase * HW + hw;
    ob[(size_t)c0 * HW]        = acc0[v] * sc;
    ob[(size_t)(c0 + 16) * HW] = acc1[v] * sc;
    ob[(size_t)(c0 + 32) * HW] = acc2[v] * sc;
    ob[(size_t)(c0 + 48) * HW] = acc3[v] * sc;
  }
}

// ---------------------------------------------------------------------------
extern "C" void kernel_launch(void* const* d_in, const int* in_sizes, int n_in,
                              void* d_out, int out_size, void* d_ws, size_t ws_size,
                              hipStream_t stream) {
  (void)in_sizes; (void)n_in; (void)out_size; (void)ws_size;
  const float* feat     = (const float*)d_in[0];
  // d_in[1] (target_mask) and d_in[4] (patch_size) provably cancel (see header).
  const float* real_mem = (const float*)d_in[2];
  const float* fake_mem = (const float*)d_in[3];
  float* out = (float*)d_out;

  // Workspace carve-up (all fp32), ~197 MB total.
  char* ws = (char*)d_ws;
  float* qf     = (float*)ws;                                  ws += (size_t)Nn * Cc * 4;
  float* score  = (float*)ws;                                  ws += (size_t)Nn * Mm * 4;
  float* qup    = (float*)ws;                                  ws += (size_t)Mm * Cc * 4;
  float* memupd = (float*)ws;                                  ws += (size_t)Mm * Cc * 4;
  float* pmax   = (float*)ws;                                  ws += (size_t)NCHUNK * Mm * 4;
  float* colmax = (float*)ws;                                  ws += (size_t)Mm * 4;
  float* rmax   = (float*)ws;                                  ws += (size_t)Nn * 4;
  float* rinv   = (float*)ws;                                  ws += (size_t)Nn * 4;

  // Shared preprocessing: qf + output qf channel blocks.
  k_qf_norm<<<Nn, 128, 0, stream>>>(feat, qf, out);

  for (int pass = 0; pass < 2; ++pass) {
    const float* mem = pass ? fake_mem : real_mem;
    const int chanBase = 512 + pass * 1024;

    k_zero<<<(Mm * Cc + 255) / 256, 256, 0, stream>>>(qup, Mm * Cc);

    dim3 g1(Nn / 16, Mm / 256);
    k_gemm_ab_t<<<g1, 128, 0, stream>>>(qf, mem, score, Cc, Mm);

    dim3 g2(Mm / 256, NCHUNK);
    k_colmax_part<<<g2, 256, 0, stream>>>(score, pmax, Nn / NCHUNK);
    k_colmax_fin<<<Mm / 256, 256, 0, stream>>>(pmax, colmax);

    k_rowpick_update<<<Nn, 256, 0, stream>>>(score, colmax, qf, qup);
    k_memupd_norm<<<Mm, 256, 0, stream>>>(qup, mem, memupd);

    k_gemm_ab_t<<<g1, 128, 0, stream>>>(qf, memupd, score, Cc, Mm);
    k_rowstats<<<Nn, 256, 0, stream>>>(score, rmax, rinv);

    dim3 g3(Nn / 16, Cc / 256);
    k_softmax_gemm_out<<<g3, 128, 0, stream>>>(score, memupd, rmax, rinv, out, chanBase);
  }
}